// SpectralLinearLayer_OFT_28028956573810
// MI455X (gfx1250) — compile-verified
//
#include <hip/hip_runtime.h>
#include <hip/hip_bf16.h>

typedef __attribute__((ext_vector_type(2))) float v2f;
typedef __attribute__((ext_vector_type(4))) float v4f;
typedef __attribute__((ext_vector_type(8))) float v8f;

#define TILE_K 32
#define PITCH  36   // LDS row pitch in floats; multiple of 4 (float4 stores), conflict-spreading

// ---------------------------------------------------------------------------
// Cayley: Q = (I - skew) @ inv(I + skew), skew = 0.5*(M - M^T), 8x8.
// Thread 0 handles spectral_A -> qa, thread 1 handles spectral_B -> qb.
// ---------------------------------------------------------------------------
__global__ void k_cayley(const float* __restrict__ A, const float* __restrict__ B,
                         float* __restrict__ qa, float* __restrict__ qb) {
  int t = threadIdx.x;
  if (t >= 2) return;
  const float* Msrc = (t == 0) ? A : B;
  float*       qout = (t == 0) ? qa : qb;
  float P[8][8], Nm[8][8], inv[8][8];
  for (int i = 0; i < 8; ++i)
    for (int j = 0; j < 8; ++j) {
      float s = 0.5f * (Msrc[i * 8 + j] - Msrc[j * 8 + i]);
      float d = (i == j) ? 1.0f : 0.0f;
      P[i][j] = d + s;      // I + skew
      Nm[i][j] = d - s;     // I - skew
      inv[i][j] = d;
    }
  // Gauss-Jordan with partial pivoting: inv <- P^-1
  for (int c = 0; c < 8; ++c) {
    int piv = c; float best = fabsf(P[c][c]);
    for (int rr = c + 1; rr < 8; ++rr) {
      float a = fabsf(P[rr][c]);
      if (a > best) { best = a; piv = rr; }
    }
    if (piv != c)
      for (int j = 0; j < 8; ++j) {
        float tp = P[c][j];  P[c][j] = P[piv][j];  P[piv][j] = tp;
        float ti = inv[c][j]; inv[c][j] = inv[piv][j]; inv[piv][j] = ti;
      }
    float r = 1.0f / P[c][c];
    for (int j = 0; j < 8; ++j) { P[c][j] *= r; inv[c][j] *= r; }
    for (int rr = 0; rr < 8; ++rr) if (rr != c) {
      float f = P[rr][c];
      for (int j = 0; j < 8; ++j) {
        P[rr][j]   -= f * P[c][j];
        inv[rr][j] -= f * inv[c][j];
      }
    }
  }
  for (int i = 0; i < 8; ++i)
    for (int j = 0; j < 8; ++j) {
      float acc = 0.0f;
      for (int k = 0; k < 8; ++k) acc += Nm[i][k] * inv[k][j];
      qout[i * 8 + j] = acc;
    }
}

// ---------------------------------------------------------------------------
// UCa[m,i] = (sum_j U[m,lo+j]*QA[j,i]) * S[lo+i]*C[i]   (A-side slice, pre-scaled)
// VCb[m,i] =  sum_j V[m,lo+j]*QB[j,i]                   (B-side slice)
// ---------------------------------------------------------------------------
__global__ void k_rank8(const float* __restrict__ U, const float* __restrict__ V,
                        const float* __restrict__ S, const float* __restrict__ C,
                        const int* __restrict__ idxp,
                        const float* __restrict__ qa, const float* __restrict__ qb,
                        float* __restrict__ UCa, float* __restrict__ VCb) {
  int m = blockIdx.x * blockDim.x + threadIdx.x;   // 0..4095
  int lo = idxp[0] * 8;
  float u[8], v[8];
#pragma unroll
  for (int j = 0; j < 8; ++j) {
    u[j] = U[(long)m * 4096 + lo + j];
    v[j] = V[(long)m * 4096 + lo + j];
  }
#pragma unroll
  for (int i = 0; i < 8; ++i) {
    float sa = 0.0f, sb = 0.0f;
#pragma unroll
    for (int j = 0; j < 8; ++j) {
      sa += u[j] * qa[j * 8 + i];
      sb += v[j] * qb[j * 8 + i];
    }
    UCa[(long)m * 8 + i] = sa * (S[lo + i] * C[i]);
    VCb[(long)m * 8 + i] = sb;
  }
}

// ---------------------------------------------------------------------------
// C[m,n] = sum_k A'[m,k] * B'[n,k]  (+ bias[n])
//   SLICE=1:  A'[m,k] = k in [lo,lo+8) ? UCa[m,k-lo] : A[m,k]*S[k]
//             B'[n,k] = k in [lo,lo+8) ? VCb[n,k-lo] : B[n,k]
//   SLICE=0:  plain A/B.  NTB: non-temporal loads for B stream.
// Block: 256 thr (8 waves), tile 128(M) x 128(N) x 32(K). Wave tile: 64x32
// = 4x2 accumulators of 16x16, driven by V_WMMA_F32_16X16X4_F32.
// Register-staged double buffer: next K-tile is fetched into VGPRs before the
// WMMA burst so HBM loads overlap compute.
// ---------------------------------------------------------------------------
template <bool SLICE, bool BIAS, bool NTB>
__global__ __launch_bounds__(256)
void k_gemm_tn(const float* __restrict__ Asrc,   // [M x K] row-major
               const float* __restrict__ Bsrc,   // [N x K] row-major
               const float* __restrict__ Sscale,
               const float* __restrict__ UCa,
               const float* __restrict__ VCb,
               const int* __restrict__ idxp,
               const float* __restrict__ bias,
               float* __restrict__ Cout,         // [M x N] row-major
               int K) {
  __shared__ float As[128 * PITCH];
  __shared__ float Bs[128 * PITCH];

  const int tid  = threadIdx.x;
  const int lane = tid & 31;
  const int wave = tid >> 5;
  const int wm   = (wave & 1) * 64;   // wave M offset in block tile
  const int wn   = (wave >> 1) * 32;  // wave N offset in block tile
  const int lhi  = lane >> 4;         // 0/1: K-pair group / M-half selector
  const int llo  = lane & 15;

  int lo = 0;
  if constexpr (SLICE) lo = idxp[0] * 8;

  const long rowA0 = (long)blockIdx.y * 128;
  const long rowB0 = (long)blockIdx.x * 128;

  // Per-thread staging slots: 4 float4 chunks of A tile + 4 of B tile.
  int srow[4], sch[4];
#pragma unroll
  for (int p = 0; p < 4; ++p) {
    int id  = p * 256 + tid;   // 0..1023
    srow[p] = id >> 3;         // 0..127
    sch[p]  = (id & 7) * 4;    // 0..28
  }

  v4f ra[4], rb[4];

  // fetch one K-tile (base kt) into registers
  auto fetch = [&](int kt) {
#pragma unroll
    for (int p = 0; p < 4; ++p) {
      int kglb = kt + sch[p];
      if constexpr (SLICE) {
        if ((unsigned)(kglb - lo) < 8u) {   // lo%8==0 -> float4 chunk fully inside slice
          ra[p] = *(const v4f*)(UCa + (rowA0 + srow[p]) * 8 + (kglb - lo));
          rb[p] = *(const v4f*)(VCb + (rowB0 + srow[p]) * 8 + (kglb - lo));
        } else {
          ra[p] = __builtin_nontemporal_load(
              (const v4f*)(Asrc + (rowA0 + srow[p]) * (long)K + kglb));
          ra[p].x *= Sscale[kglb + 0]; ra[p].y *= Sscale[kglb + 1];
          ra[p].z *= Sscale[kglb + 2]; ra[p].w *= Sscale[kglb + 3];
          rb[p] = __builtin_nontemporal_load(
              (const v4f*)(Bsrc + (rowB0 + srow[p]) * (long)K + kglb));
        }
      } else {
        // A stream (x) is single-pass -> NT; B (W) must stay L2-resident.
        ra[p] = __builtin_nontemporal_load(
            (const v4f*)(Asrc + (rowA0 + srow[p]) * (long)K + kglb));
        if constexpr (NTB)
          rb[p] = __builtin_nontemporal_load(
              (const v4f*)(Bsrc + (rowB0 + srow[p]) * (long)K + kglb));
        else
          rb[p] = *(const v4f*)(Bsrc + (rowB0 + srow[p]) * (long)K + kglb);
      }
    }
  };

  v8f acc[4][2];
#pragma unroll
  for (int mt = 0; mt < 4; ++mt)
#pragma unroll
    for (int nt = 0; nt < 2; ++nt) acc[mt][nt] = (v8f)(0.0f);

  fetch(0);

  for (int kt = 0; kt < K; kt += TILE_K) {
    // publish staged registers to LDS
#pragma unroll
    for (int p = 0; p < 4; ++p) {
      *(v4f*)(As + srow[p] * PITCH + sch[p]) = ra[p];
      *(v4f*)(Bs + srow[p] * PITCH + sch[p]) = rb[p];
    }
    __syncthreads();

    // start HBM loads for the NEXT tile; they retire during the WMMA burst
    if (kt + TILE_K < K) fetch(kt + TILE_K);

    // ---- 8 K-steps of 4, 8 WMMAs each => 64 v_wmma per K-tile ----
#pragma unroll
    for (int k0 = 0; k0 < TILE_K; k0 += 4) {
      const int kk = k0 + lhi * 2;  // this lane-group's K pair base
      v2f afrag[4], bfrag[2];
#pragma unroll
      for (int mt = 0; mt < 4; ++mt)
        afrag[mt] = *(const v2f*)(As + (wm + mt * 16 + llo) * PITCH + kk);
#pragma unroll
      for (int nt = 0; nt < 2; ++nt)
        bfrag[nt] = *(const v2f*)(Bs + (wn + nt * 16 + llo) * PITCH + kk);
#pragma unroll
      for (int mt = 0; mt < 4; ++mt)
#pragma unroll
        for (int nt = 0; nt < 2; ++nt)
          acc[mt][nt] = __builtin_amdgcn_wmma_f32_16x16x4_f32(
              false, afrag[mt], false, bfrag[nt],
              (short)0, acc[mt][nt], false, false);
    }
    __syncthreads();
  }

  // ---- epilogue: C/D layout -> row = v + 8*lhi (+16*mt), col = llo (+16*nt) ----
  const long Ncols = (long)gridDim.x * 128;
#pragma unroll
  for (int mt = 0; mt < 4; ++mt)
#pragma unroll
    for (int nt = 0; nt < 2; ++nt) {
      long gcol = rowB0 + wn + nt * 16 + llo;
      float badd = 0.0f;
      if constexpr (BIAS) badd = bias[gcol];
#pragma unroll
      for (int v = 0; v < 8; ++v) {
        long grow = rowA0 + wm + mt * 16 + lhi * 8 + v;
        float val = acc[mt][nt][v] + badd;
        if constexpr (BIAS)   // final output: write-once, keep it out of L2
          __builtin_nontemporal_store(val, Cout + grow * Ncols + gcol);
        else                  // W: keep resident for the next GEMM
          Cout[grow * Ncols + gcol] = val;
      }
    }
}

// ---------------------------------------------------------------------------
extern "C" void kernel_launch(void* const* d_in, const int* in_sizes, int n_in,
                              void* d_out, int out_size, void* d_ws, size_t ws_size,
                              hipStream_t stream) {
  const float* x    = (const float*)d_in[0];  // [4,2048,4096]
  const float* U    = (const float*)d_in[1];  // [4096,4096]
  const float* S    = (const float*)d_in[2];  // [4096]
  const float* V    = (const float*)d_in[3];  // [4096,4096]
  const float* spA  = (const float*)d_in[4];  // [8,8]
  const float* spB  = (const float*)d_in[5];  // [8,8]
  const float* spC  = (const float*)d_in[6];  // [8]
  const float* bias = (const float*)d_in[7];  // [4096]
  const int*   idxp = (const int*)d_in[8];    // scalar
  (void)in_sizes; (void)n_in; (void)ws_size;

  char*  ws  = (char*)d_ws;
  float* qa  = (float*)(ws + 0);
  float* qb  = (float*)(ws + 256);
  float* UCa = (float*)(ws + 1024);                    // 4096 x 8
  float* VCb = (float*)(ws + 1024 + 4096 * 8 * 4);     // 4096 x 8
  float* W   = (float*)(ws + (1 << 20));               // 4096 x 4096 (64 MiB, L2-resident)

  k_cayley<<<1, 64, 0, stream>>>(spA, spB, qa, qb);
  k_rank8<<<4096 / 256, 256, 0, stream>>>(U, V, S, spC, idxp, qa, qb, UCa, VCb);

  // W[m(out), n(in)] = sum_k (U[m,k]*S[k]) * V[n,k], with rank-8 slice swap.
  k_gemm_tn<true, false, true><<<dim3(4096 / 128, 4096 / 128), 256, 0, stream>>>(
      U, V, S, UCa, VCb, idxp, nullptr, W, 4096);

  // out[m, o] = sum_k x[m,k] * W[o,k] + bias[o]
  k_gemm_tn<false, true, false><<<dim3(4096 / 128, 8192 / 128), 256, 0, stream>>>(
      x, W, nullptr, nullptr, nullptr, nullptr, bias, (float*)d_out, 4096);
}